// Net_73349451481424
// MI455X (gfx1250) — compile-verified
//
#include <hip/hip_runtime.h>
#include <hip/hip_bf16.h>

#define NN     100000
#define DIN    512
#define DHID   16
#define DOUT   64
#define NTILES (NN / 16)   // 6250, exact

typedef __attribute__((ext_vector_type(2))) float v2f;
typedef __attribute__((ext_vector_type(8))) float v8f;

__device__ __forceinline__ v8f wmma_f32_k4(v2f a, v2f b, v8f c) {
#if defined(__HIP_DEVICE_COMPILE__)
#if __has_builtin(__builtin_amdgcn_wmma_f32_16x16x4_f32)
    // (neg_a, A, neg_b, B, c_mod, C, reuse_a, reuse_b)
    return __builtin_amdgcn_wmma_f32_16x16x4_f32(false, a, false, b, (short)0, c,
                                                 false, false);
#else
#error "wmma_f32_16x16x4_f32 builtin not available on DEVICE toolchain"
#endif
#else
    // host pass stub; device code never runs on host
    (void)a; (void)b;
    return c;
#endif
}

// ---------------------------------------------------------------- init / degree
__global__ void k_init(float* __restrict__ deg, float* __restrict__ agg1,
                       float* __restrict__ out) {
    int i = blockIdx.x * blockDim.x + threadIdx.x;
    if (i < NN) deg[i] = 1.0f;                 // self-loop contributes 1 to degree
    if (i < NN * DHID) agg1[i] = 0.0f;
    if (i < NN * DOUT) out[i] = 0.0f;
}

__global__ void k_degree(const int* __restrict__ ei, float* __restrict__ deg,
                         long long E) {
    long long e = (long long)blockIdx.x * blockDim.x + threadIdx.x;
    if (e >= E) return;
    int d = ei[E + e];                          // edge_index[1][e]
    atomicAdd(&deg[d], 1.0f);
}

__global__ void k_rsqrt(float* __restrict__ deg) {
    int i = blockIdx.x * blockDim.x + threadIdx.x;
    if (i < NN) deg[i] = __frsqrt_rn(deg[i]);   // deg >= 1, no zero guard needed
}

// ---------------------------------------------------------------- GEMM1: h = x @ W1
// one wave32 per 16-row tile; W1 (512x16 f32 = 32KB) staged in LDS
__global__ __launch_bounds__(128) void k_gemm1(const float* __restrict__ x,
                                               const float* __restrict__ W1,
                                               float* __restrict__ h) {
    __shared__ float lw[DIN * DHID];
    for (int i = threadIdx.x; i < DIN * DHID; i += 128) lw[i] = W1[i];
    __syncthreads();

    int wave = threadIdx.x >> 5, lane = threadIdx.x & 31;
    int tile = blockIdx.x * 4 + wave;
    if (tile >= NTILES) return;                 // wave-uniform: EXEC stays all-1s
    int row0 = tile * 16;
    int m    = lane & 15;
    int khi  = (lane >> 4) << 1;                // lanes 0-15 -> K+0/1, 16-31 -> K+2/3

    const float* arow = x + (size_t)(row0 + m) * DIN + khi;
    v8f c = {};
    for (int k = 0; k < DIN; k += 4) {
        v2f a = *(const v2f*)(arow + k);        // A[m][k+khi], A[m][k+khi+1]
        v2f b;
        b.x = lw[(k + khi) * DHID + m];         // B[k+khi][m]
        b.y = lw[(k + khi + 1) * DHID + m];
        c = wmma_f32_k4(a, b, c);
    }
    int rbase = row0 + ((lane >> 4) << 3);      // VGPR r -> M=r (lo) / M=8+r (hi)
#pragma unroll
    for (int r = 0; r < 8; ++r)
        h[(size_t)(rbase + r) * DHID + m] = c[r];
}

// ---------------------------------------------------------------- edge scatter L1
// thread = (edge, feature); h/agg1 are L2-resident (6.4MB each)
__global__ void k_scatter1(const int* __restrict__ ei, const float* __restrict__ dis,
                           const float* __restrict__ h, float* __restrict__ agg1,
                           long long E) {
    long long t = (long long)blockIdx.x * blockDim.x + threadIdx.x;
    long long e = t >> 4;
    int j = (int)(t & 15);
    if (e >= E) return;
    int s = ei[e], d = ei[E + e];
    float w = dis[s] * dis[d];
    atomicAdd(&agg1[(size_t)d * DHID + j], h[(size_t)s * DHID + j] * w);
}

// ---------------------------------------------------------------- GEMM2
// h2 = relu(agg1 + dis^2*h + b1) @ W2   (self-loop + bias + relu fused into A-load)
__global__ __launch_bounds__(128) void k_gemm2(const float* __restrict__ agg1,
                                               const float* __restrict__ h,
                                               const float* __restrict__ dis,
                                               const float* __restrict__ b1,
                                               const float* __restrict__ W2,
                                               float* __restrict__ h2) {
    __shared__ float lw[DHID * DOUT];           // 4KB
    for (int i = threadIdx.x; i < DHID * DOUT; i += 128) lw[i] = W2[i];
    __syncthreads();

    int wave = threadIdx.x >> 5, lane = threadIdx.x & 31;
    int tile = blockIdx.x * 4 + wave;
    if (tile >= NTILES) return;
    int row0 = tile * 16;
    int m    = lane & 15;
    int khi  = (lane >> 4) << 1;
    int row  = row0 + m;
    float ds2 = dis[row] * dis[row];
    const float* ag = agg1 + (size_t)row * DHID;
    const float* hr = h    + (size_t)row * DHID;

    v8f c0 = {}, c1 = {}, c2 = {}, c3 = {};
#pragma unroll
    for (int kk = 0; kk < DHID; kk += 4) {
        int k0 = kk + khi;
        v2f a;
        a.x = fmaxf(ag[k0]     + ds2 * hr[k0]     + b1[k0],     0.0f);
        a.y = fmaxf(ag[k0 + 1] + ds2 * hr[k0 + 1] + b1[k0 + 1], 0.0f);
        v2f b;
        b.x = lw[k0 * DOUT + m];        b.y = lw[(k0 + 1) * DOUT + m];
        c0 = wmma_f32_k4(a, b, c0);
        b.x = lw[k0 * DOUT + 16 + m];   b.y = lw[(k0 + 1) * DOUT + 16 + m];
        c1 = wmma_f32_k4(a, b, c1);
        b.x = lw[k0 * DOUT + 32 + m];   b.y = lw[(k0 + 1) * DOUT + 32 + m];
        c2 = wmma_f32_k4(a, b, c2);
        b.x = lw[k0 * DOUT + 48 + m];   b.y = lw[(k0 + 1) * DOUT + 48 + m];
        c3 = wmma_f32_k4(a, b, c3);
    }
    int rbase = row0 + ((lane >> 4) << 3);
#pragma unroll
    for (int r = 0; r < 8; ++r) {
        float* o = h2 + (size_t)(rbase + r) * DOUT + m;
        o[0]  = c0[r];
        o[16] = c1[r];
        o[32] = c2[r];
        o[48] = c3[r];
    }
}

// ---------------------------------------------------------------- edge scatter L2
__global__ void k_scatter2(const int* __restrict__ ei, const float* __restrict__ dis,
                           const float* __restrict__ h2, float* __restrict__ out,
                           long long E) {
    long long t = (long long)blockIdx.x * blockDim.x + threadIdx.x;
    long long e = t >> 6;
    int j = (int)(t & 63);
    if (e >= E) return;
    int s = ei[e], d = ei[E + e];
    float w = dis[s] * dis[d];
    atomicAdd(&out[(size_t)d * DOUT + j], h2[(size_t)s * DOUT + j] * w);
}

// ---------------------------------------------------------------- log_softmax
// one wave per node; fuse self-loop + bias; wave32 shfl reductions, in-place on out
__global__ __launch_bounds__(256) void k_lsm(const float* __restrict__ h2,
                                             const float* __restrict__ dis,
                                             const float* __restrict__ b2,
                                             float* __restrict__ out) {
    int wave = threadIdx.x >> 5, lane = threadIdx.x & 31;
    int node = blockIdx.x * 8 + wave;
    if (node >= NN) return;
    float w2 = dis[node] * dis[node];
    size_t base = (size_t)node * DOUT;
    float v0 = out[base + lane]      + w2 * h2[base + lane]      + b2[lane];
    float v1 = out[base + lane + 32] + w2 * h2[base + lane + 32] + b2[lane + 32];
    float mx = fmaxf(v0, v1);
#pragma unroll
    for (int off = 16; off; off >>= 1) mx = fmaxf(mx, __shfl_xor(mx, off, 32));
    float s = __expf(v0 - mx) + __expf(v1 - mx);
#pragma unroll
    for (int off = 16; off; off >>= 1) s += __shfl_xor(s, off, 32);
    float lse = mx + __logf(s);
    out[base + lane]      = v0 - lse;
    out[base + lane + 32] = v1 - lse;
}

// ----------------------------------------------------------------
extern "C" void kernel_launch(void* const* d_in, const int* in_sizes, int n_in,
                              void* d_out, int out_size, void* d_ws, size_t ws_size,
                              hipStream_t stream) {
    const float* x  = (const float*)d_in[0];
    const int*   ei = (const int*)d_in[1];       // edge_index [2, E]
    const float* W1 = (const float*)d_in[2];
    const float* b1 = (const float*)d_in[3];
    const float* W2 = (const float*)d_in[4];
    const float* b2 = (const float*)d_in[5];
    float* out = (float*)d_out;
    long long E = (long long)in_sizes[1] / 2;

    float* ws   = (float*)d_ws;
    float* dis  = ws;                              // NN        (deg -> rsqrt in place)
    float* h    = dis  + NN;                       // NN*16
    float* agg1 = h    + (size_t)NN * DHID;        // NN*16
    float* h2   = agg1 + (size_t)NN * DHID;        // NN*64

    k_init  <<<(NN * DOUT + 255) / 256, 256, 0, stream>>>(dis, agg1, out);
    k_degree<<<(unsigned)((E + 255) / 256), 256, 0, stream>>>(ei, dis, E);
    k_rsqrt <<<(NN + 255) / 256, 256, 0, stream>>>(dis);

    k_gemm1 <<<(NTILES + 3) / 4, 128, 0, stream>>>(x, W1, h);
    k_scatter1<<<(unsigned)((E * DHID + 255) / 256), 256, 0, stream>>>(ei, dis, h, agg1, E);
    k_gemm2 <<<(NTILES + 3) / 4, 128, 0, stream>>>(agg1, h, dis, b1, W2, h2);
    k_scatter2<<<(unsigned)((E * DOUT + 255) / 256), 256, 0, stream>>>(ei, dis, h2, out, E);
    k_lsm   <<<(NN + 7) / 8, 256, 0, stream>>>(h2, dis, b2, out);
}